// BilateralFilter_55637006352450
// MI455X (gfx1250) — compile-verified
//
#include <hip/hip_runtime.h>
#include <cstdint>

#define HH   320
#define WW   320
#define NB   2
#define CIN  3
#define NCLS 6
#define TILE 16
#define HALO 2
#define TD   (TILE + 2 * HALO)   // 20
#define NPIX (TD * TD)           // 400

typedef float v2f __attribute__((ext_vector_type(2)));

__launch_bounds__(256)
__global__ void crf_bilateral_kernel(const float* __restrict__ Q,
                                     const float* __restrict__ I,
                                     float* __restrict__ out)
{
    // I tile: planar [cin][y][x]  (stride-1 b32 reads, conflict-free)
    __shared__ float sI[CIN][TD][TD];          // 1200 floats
    // Q tile: per-pixel AoS [y][x][cls] (stride-6 dwords; 8B-aligned float2 pairs)
    __shared__ float sQa[NPIX * NCLS];         // 2400 floats

    const int b   = blockIdx.z;
    const int ty0 = blockIdx.y * TILE - HALO;
    const int tx0 = blockIdx.x * TILE - HALO;
    const int tid = threadIdx.y * TILE + threadIdx.x;

    const float* __restrict__ Ibase = I + (size_t)b * CIN * HH * WW;
    const float* __restrict__ Qbase = Q + (size_t)b * NCLS * HH * WW;

    const int NI   = CIN * NPIX;           // 1200
    const int NTOT = NI + NCLS * NPIX;     // 3600

    // Stage halo tiles into LDS with CDNA5 async-to-LDS loads (ASYNCcnt path).
    // Per-lane LDS destinations make the planar->AoS transpose of the Q tile
    // free (done inside the async data mover) while global reads stay coalesced.
    for (int e = tid; e < NTOT; e += 256) {
        const bool isI = e < NI;
        const int  r   = isI ? e : (e - NI);
        const int  ch  = r / NPIX;                  // cin or cls
        const int  p   = r - ch * NPIX;             // pixel index in tile
        const int  yy  = p / TD;
        const int  xx  = p - yy * TD;
        const int  gy  = ty0 + yy;
        const int  gx  = tx0 + xx;
        float* ldsPtr  = isI ? (&sI[0][0][0] + ch * NPIX + p)
                             : (&sQa[0] + p * NCLS + ch);
        if ((unsigned)gy < (unsigned)HH && (unsigned)gx < (unsigned)WW) {
            const float* gb = isI ? Ibase : Qbase;
            const unsigned voff  = (unsigned)(((ch * HH + gy) * WW + gx) * 4);
            const unsigned laddr = (unsigned)(uintptr_t)ldsPtr; // low 32 bits of flat LDS ptr
            asm volatile("global_load_async_to_lds_b32 %0, %1, %2"
                         :
                         : "v"(laddr), "v"(voff), "s"(gb)
                         : "memory");
        } else {
            *ldsPtr = 0.0f;   // reference zero-pads; Ki is evaluated vs these zeros
        }
    }
    asm volatile("s_wait_asynccnt 0" ::: "memory");
    __syncthreads();

    const int lx = threadIdx.x + HALO;
    const int ly = threadIdx.y + HALO;

    // log2(spatial weight) = -(dx^2+dy^2)/8 * log2(e) = -0.18033688011 * s
    // (folded into the exp2 below: Ki*w = exp2(-72.134752*d^2 + L2W))
    constexpr float L2W[5][5] = {
        {-1.44269504f, -0.90168440f, -0.72134752f, -0.90168440f, -1.44269504f},
        {-0.90168440f, -0.36067376f, -0.18033688f, -0.36067376f, -0.90168440f},
        {-0.72134752f, -0.18033688f,  0.00000000f, -0.18033688f, -0.72134752f},
        {-0.90168440f, -0.36067376f, -0.18033688f, -0.36067376f, -0.90168440f},
        {-1.44269504f, -0.90168440f, -0.72134752f, -0.90168440f, -1.44269504f}};
    constexpr float KNEG50L2E = -72.13475204444817f;  // -50 * log2(e)

    const float Ic0 = sI[0][ly][lx];
    const float Ic1 = sI[1][ly][lx];
    const float Ic2 = sI[2][ly][lx];

    // acc[cin][pair]: pair j covers classes {2j, 2j+1}; forced v_pk_fma_f32
    v2f acc[CIN][NCLS / 2];
#pragma unroll
    for (int ci = 0; ci < CIN; ++ci)
#pragma unroll
        for (int j = 0; j < NCLS / 2; ++j) acc[ci][j] = (v2f){0.0f, 0.0f};
    float nrm[CIN] = {};

#pragma unroll
    for (int dy = -2; dy <= 2; ++dy) {
#pragma unroll
        for (int dx = -2; dx <= 2; ++dx) {
            if (dx == 0 && dy == 0) continue;
            const float l2w = L2W[dy + 2][dx + 2];
            const int yy = ly + dy, xx = lx + dx;
            const float d0 = sI[0][yy][xx] - Ic0;
            const float d1 = sI[1][yy][xx] - Ic1;
            const float d2 = sI[2][yy][xx] - Ic2;
            // a = Ki * w_k = exp2(d^2 * (-50*log2e) + log2(w_k))
            const float a0 = __builtin_amdgcn_exp2f(__builtin_fmaf(d0 * d0, KNEG50L2E, l2w));
            const float a1 = __builtin_amdgcn_exp2f(__builtin_fmaf(d1 * d1, KNEG50L2E, l2w));
            const float a2 = __builtin_amdgcn_exp2f(__builtin_fmaf(d2 * d2, KNEG50L2E, l2w));
            nrm[0] += a0; nrm[1] += a1; nrm[2] += a2;

            const v2f a0v = (v2f){a0, a0};
            const v2f a1v = (v2f){a1, a1};
            const v2f a2v = (v2f){a2, a2};
            const int p = yy * TD + xx;
            const v2f* qp = (const v2f*)&sQa[p * NCLS]; // 24B stride -> 8B aligned
#pragma unroll
            for (int j = 0; j < NCLS / 2; ++j) {
                const v2f q = qp[j];                     // ds_load_b64
                // D = S0*S1 + S2  (CDNA5 VOP3P packed f32: 2 FMAs per slot)
                asm("v_pk_fma_f32 %0, %1, %2, %0" : "+v"(acc[0][j]) : "v"(a0v), "v"(q));
                asm("v_pk_fma_f32 %0, %1, %2, %0" : "+v"(acc[1][j]) : "v"(a1v), "v"(q));
                asm("v_pk_fma_f32 %0, %1, %2, %0" : "+v"(acc[2][j]) : "v"(a2v), "v"(q));
            }
        }
    }

    const int gy = blockIdx.y * TILE + threadIdx.y;
    const int gx = blockIdx.x * TILE + threadIdx.x;
    const size_t HW  = (size_t)HH * WW;
    const size_t pix = (size_t)gy * WW + gx;
#pragma unroll
    for (int ci = 0; ci < CIN; ++ci) {
        const float rn = 1.0f / nrm[ci];
#pragma unroll
        for (int j = 0; j < NCLS / 2; ++j) {
            out[(((size_t)b * CIN + ci) * NCLS + 2 * j + 0) * HW + pix] = acc[ci][j].x * rn;
            out[(((size_t)b * CIN + ci) * NCLS + 2 * j + 1) * HW + pix] = acc[ci][j].y * rn;
        }
    }
}

extern "C" void kernel_launch(void* const* d_in, const int* in_sizes, int n_in,
                              void* d_out, int out_size, void* d_ws, size_t ws_size,
                              hipStream_t stream) {
    const float* Q = (const float*)d_in[0];   // (2,6,320,320) f32
    const float* I = (const float*)d_in[1];   // (2,3,320,320) f32
    float* out = (float*)d_out;               // (2,3,6,320,320) f32
    dim3 grid(WW / TILE, HH / TILE, NB);      // (20,20,2)
    dim3 block(TILE, TILE, 1);                // 256 threads = 8 wave32
    hipLaunchKernelGGL(crf_bilateral_kernel, grid, block, 0, stream, Q, I, out);
}